// GCNModel_89541478187016
// MI455X (gfx1250) — compile-verified
//
#include <hip/hip_runtime.h>
#include <hip/hip_bf16.h>
#include <math.h>

typedef __attribute__((ext_vector_type(2))) float v2f;
typedef __attribute__((ext_vector_type(8))) float v8f;
typedef int v4i_vs __attribute__((vector_size(16)));   // matches builtin param type

#define NNODES 15135
#define BS 8
#define HFEAT 128
#define NEDGE 242160

// ---------------------------------------------------------------------------
// Degree / normalization
// ---------------------------------------------------------------------------
__global__ void deg_init_kernel(float* deg, int n) {
    int i = blockIdx.x * blockDim.x + threadIdx.x;
    if (i < n) deg[i] = 1.0f;  // self loop
}

__global__ void deg_edge_kernel(const int* __restrict__ ei, float* deg, int E) {
    int e = blockIdx.x * blockDim.x + threadIdx.x;
    if (e < E) atomicAdd(&deg[ei[E + e]], 1.0f);  // dst row of edge_index
}

__global__ void deg_rsqrt_kernel(float* deg, int n) {
    int i = blockIdx.x * blockDim.x + threadIdx.x;
    if (i < n) deg[i] = rsqrtf(deg[i]);  // becomes dinv
}

// ---------------------------------------------------------------------------
// WMMA f32 GEMM: Y[M,128] = X[M,K] * W[K,128]   (K = 64 or 128)
// block: 256 threads (8 waves). Each block: 32 rows x 128 cols.
// Wave w handles columns [16w, 16w+16) with TWO 16x16 accumulators sharing
// each B fragment (one W load feeds two v_wmma_f32_16x16x4_f32 ops).
// X tile (32 x K) staged in LDS via async global->LDS copies when available.
// ---------------------------------------------------------------------------
__global__ void gemm_wmma_f32(const float* __restrict__ X,
                              const float* __restrict__ W,
                              float* __restrict__ Y,
                              int M, int K) {
    __shared__ float sX[32 * 128];     // 16 KB max
    const int tid   = threadIdx.x;
    const int wave  = tid >> 5;
    const int lane  = tid & 31;
    const int tileM = blockIdx.x * 32;

    // ---- stage 32 x K tile of X into LDS (16B vectors) ----
    const int nvec = (32 * K) >> 2;    // float4 count
#if defined(__gfx1250__) && __has_builtin(__builtin_amdgcn_global_load_async_to_lds_b128)
    for (int i = tid; i < nvec; i += 256) {
        int r  = i / (K >> 2);
        int c4 = (i % (K >> 2)) << 2;
        int gr = tileM + r;
        if (gr < M) {
            __builtin_amdgcn_global_load_async_to_lds_b128(
                (__attribute__((address_space(1))) v4i_vs*)(X + (size_t)gr * K + c4),
                (__attribute__((address_space(3))) v4i_vs*)&sX[r * K + c4],
                /*offset=*/0, /*cpol=*/0);
        } else {
            float4 zq = make_float4(0.f, 0.f, 0.f, 0.f);
            *(float4*)&sX[r * K + c4] = zq;
        }
    }
#if __has_builtin(__builtin_amdgcn_s_wait_asynccnt)
    __builtin_amdgcn_s_wait_asynccnt(0);
#else
    asm volatile("s_wait_asynccnt 0x0" ::: "memory");
#endif
#else
    for (int i = tid; i < nvec; i += 256) {
        int r  = i / (K >> 2);
        int c4 = (i % (K >> 2)) << 2;
        int gr = tileM + r;
        float4 v = make_float4(0.f, 0.f, 0.f, 0.f);
        if (gr < M) v = *(const float4*)(X + (size_t)gr * K + c4);
        *(float4*)&sX[r * K + c4] = v;
    }
#endif
    __syncthreads();

    const int n0   = wave * 16;        // column base for this wave
    const int idx  = lane & 15;        // A row within tile / B column
    const int kOff = (lane >> 4) * 2;  // lanes 16..31 hold K+2,K+3

    v8f acc0 = {};                     // rows tileM .. tileM+15
    v8f acc1 = {};                     // rows tileM+16 .. tileM+31
    for (int k0 = 0; k0 < K; k0 += 4) {
        v2f b;
        const float* wp = W + (size_t)(k0 + kOff) * HFEAT + n0 + idx;
        b.x = wp[0];
        b.y = wp[HFEAT];
        v2f a0, a1;
        a0.x = sX[idx * K + k0 + kOff];
        a0.y = sX[idx * K + k0 + kOff + 1];
        a1.x = sX[(idx + 16) * K + k0 + kOff];
        a1.y = sX[(idx + 16) * K + k0 + kOff + 1];
        acc0 = __builtin_amdgcn_wmma_f32_16x16x4_f32(
            false, a0, false, b, (short)0, acc0, false, false);
        acc1 = __builtin_amdgcn_wmma_f32_16x16x4_f32(
            false, a1, false, b, (short)0, acc1, false, false);
    }

    // C/D layout: VGPR r -> M = r + 8*(lane>=16), N = lane&15
    const int half = lane >> 4;
    const int col  = n0 + idx;
#pragma unroll
    for (int r = 0; r < 8; ++r) {
        int row0 = tileM + r + 8 * half;
        if (row0 < M) Y[(size_t)row0 * HFEAT + col] = acc0[r];
        int row1 = tileM + 16 + r + 8 * half;
        if (row1 < M) Y[(size_t)row1 * HFEAT + col] = acc1[r];
    }
}

// ---------------------------------------------------------------------------
// Aggregation: out = scatter_add(norm * XL[src]) including self loops
// ---------------------------------------------------------------------------
__global__ void agg_self_kernel(const float* __restrict__ XL,
                                const float* __restrict__ dinv,
                                float* __restrict__ out, int total) {
    int i = blockIdx.x * blockDim.x + threadIdx.x;
    if (i >= total) return;
    int n = (i >> 7) % NNODES;      // feature dim = 128
    float d = dinv[n];
    out[i] = XL[i] * d * d;         // self-loop norm = dinv[n]^2
}

__global__ void agg_edge_kernel(const float* __restrict__ XL,
                                const int* __restrict__ ei,
                                const float* __restrict__ dinv,
                                float* __restrict__ out) {
    const int e = blockIdx.x;
    const int b = blockIdx.y;
    const int f = threadIdx.x;      // 128 threads
    const int s = ei[e];
    const int d = ei[NEDGE + e];
    const float nrm = dinv[s] * dinv[d];
    const float v = XL[((size_t)b * NNODES + s) * HFEAT + f] * nrm;
    atomicAdd(out + ((size_t)b * NNODES + d) * HFEAT + f, v);
}

__global__ void bias_relu_kernel(float* __restrict__ h,
                                 const float* __restrict__ bias, int total) {
    int i = blockIdx.x * blockDim.x + threadIdx.x;
    if (i < total) h[i] = fmaxf(h[i] + bias[i & (HFEAT - 1)], 0.0f);
}

// ---------------------------------------------------------------------------
// g accumulation:  g[b,n] += sum_k h[b,n,k] * Wfc[3k + l]
// One wave per (b,n) node; lanes stride the 128 features.
// ---------------------------------------------------------------------------
__global__ void g_init_kernel(const float* __restrict__ bfc, float* g, int total) {
    int i = blockIdx.x * blockDim.x + threadIdx.x;
    if (i < total) g[i] = bfc[0];
}

__global__ void g_acc_kernel(const float* __restrict__ h,
                             const float* __restrict__ Wfc,
                             float* __restrict__ g, int total, int l) {
    int wid  = (blockIdx.x * blockDim.x + threadIdx.x) >> 5;
    int lane = threadIdx.x & 31;
    if (wid >= total) return;
    const float* row = h + (size_t)wid * HFEAT;
    float s = 0.0f;
#pragma unroll
    for (int k = lane; k < HFEAT; k += 32) s += row[k] * Wfc[3 * k + l];
#pragma unroll
    for (int off = 16; off > 0; off >>= 1) s += __shfl_down(s, off, 32);
    if (lane == 0) g[wid] += s;
}

// ---------------------------------------------------------------------------
// z = relu(g @ Wl1 + bl1):  [8,N] x [N,512]
// ---------------------------------------------------------------------------
#define ZCHUNK 253  // ceil(15135/60)
__global__ void z_init_kernel(const float* __restrict__ bl1, float* z) {
    int i = blockIdx.x * blockDim.x + threadIdx.x;
    if (i < BS * 512) z[i] = bl1[i & 511];
}

__global__ void z_acc_kernel(const float* __restrict__ g,
                             const float* __restrict__ Wl1,
                             float* __restrict__ z) {
    const int b = blockIdx.x;
    const int n0 = blockIdx.y * ZCHUNK;
    const int n1 = min(n0 + ZCHUNK, NNODES);
    const int o = threadIdx.x;      // 512 threads, coalesced over Wl1 columns
    float s = 0.0f;
    for (int n = n0; n < n1; ++n)
        s += g[b * NNODES + n] * Wl1[(size_t)n * 512 + o];
    atomicAdd(&z[b * 512 + o], s);
}

__global__ void z_relu_kernel(float* z) {
    int i = blockIdx.x * blockDim.x + threadIdx.x;
    if (i < BS * 512) z[i] = fmaxf(z[i], 0.0f);
}

// ---------------------------------------------------------------------------
// logits = z @ Wl2 + bl2, then log_softmax. One block per batch.
// ---------------------------------------------------------------------------
__global__ void head_kernel(const float* __restrict__ z,
                            const float* __restrict__ Wl2,
                            const float* __restrict__ bl2,
                            float* __restrict__ out) {
    const int b = blockIdx.x;
    const int t = threadIdx.x;      // 512
    __shared__ float sz[512];
    __shared__ float logit[10];
    sz[t] = z[b * 512 + t];
    __syncthreads();
    if (t < 10) {
        float s = bl2[t];
        for (int k = 0; k < 512; ++k) s += sz[k] * Wl2[k * 10 + t];
        logit[t] = s;
    }
    __syncthreads();
    if (t == 0) {
        float m = logit[0];
        for (int c = 1; c < 10; ++c) m = fmaxf(m, logit[c]);
        float se = 0.0f;
        for (int c = 0; c < 10; ++c) se += expf(logit[c] - m);
        float lse = m + logf(se);
        for (int c = 0; c < 10; ++c) out[b * 10 + c] = logit[c] - lse;
    }
}

// ---------------------------------------------------------------------------
// Orchestration
// ---------------------------------------------------------------------------
extern "C" void kernel_launch(void* const* d_in, const int* in_sizes, int n_in,
                              void* d_out, int out_size, void* d_ws, size_t ws_size,
                              hipStream_t stream) {
    const float* x    = (const float*)d_in[0];
    const int*   ei   = (const int*)  d_in[1];
    const float* W1   = (const float*)d_in[2];
    const float* b1   = (const float*)d_in[3];
    const float* W2   = (const float*)d_in[4];
    const float* b2   = (const float*)d_in[5];
    const float* W3   = (const float*)d_in[6];
    const float* b3   = (const float*)d_in[7];
    const float* Wfc  = (const float*)d_in[8];
    const float* bfc  = (const float*)d_in[9];
    const float* Wl1  = (const float*)d_in[10];
    const float* bl1  = (const float*)d_in[11];
    const float* Wl2  = (const float*)d_in[12];
    const float* bl2  = (const float*)d_in[13];
    float* out = (float*)d_out;

    // workspace layout
    char* ws = (char*)d_ws;
    size_t off = 0;
    auto carve = [&](size_t bytes) {
        char* p = ws + off;
        off += (bytes + 255) & ~(size_t)255;
        return p;
    };
    const size_t featBytes = (size_t)BS * NNODES * HFEAT * sizeof(float);
    float* dinv = (float*)carve((size_t)NNODES * sizeof(float));       // deg -> dinv in place
    float* g    = (float*)carve((size_t)BS * NNODES * sizeof(float));
    float* z    = (float*)carve((size_t)BS * 512 * sizeof(float));
    float* T    = (float*)carve(featBytes);   // pre-aggregation GEMM output
    float* hA   = (float*)carve(featBytes);   // ping
    float* hB   = (float*)carve(featBytes);   // pong

    const int M = BS * NNODES;                 // 121080 rows
    const int total = M * HFEAT;               // 15.5M elems
    const dim3 blk256(256);

    // --- normalization ---
    deg_init_kernel<<<dim3((NNODES + 255) / 256), blk256, 0, stream>>>(dinv, NNODES);
    deg_edge_kernel<<<dim3((NEDGE + 255) / 256), blk256, 0, stream>>>(ei, dinv, NEDGE);
    deg_rsqrt_kernel<<<dim3((NNODES + 255) / 256), blk256, 0, stream>>>(dinv, NNODES);

    g_init_kernel<<<dim3((M + 255) / 256), blk256, 0, stream>>>(bfc, g, M);

    const dim3 gemmGrid((M + 31) / 32);
    const dim3 aggGrid((total + 255) / 256);
    const dim3 edgeGrid(NEDGE, BS);
    const dim3 gaccGrid(M / 8);                // 8 waves (nodes) per 256-thread block

    // --- layer 1: x(K=64) -> hA ---
    gemm_wmma_f32<<<gemmGrid, blk256, 0, stream>>>(x, W1, T, M, 64);
    agg_self_kernel<<<aggGrid, blk256, 0, stream>>>(T, dinv, hA, total);
    agg_edge_kernel<<<edgeGrid, dim3(HFEAT), 0, stream>>>(T, ei, dinv, hA);
    bias_relu_kernel<<<aggGrid, blk256, 0, stream>>>(hA, b1, total);
    g_acc_kernel<<<gaccGrid, blk256, 0, stream>>>(hA, Wfc, g, M, 0);

    // --- layer 2: hA(K=128) -> hB ---
    gemm_wmma_f32<<<gemmGrid, blk256, 0, stream>>>(hA, W2, T, M, 128);
    agg_self_kernel<<<aggGrid, blk256, 0, stream>>>(T, dinv, hB, total);
    agg_edge_kernel<<<edgeGrid, dim3(HFEAT), 0, stream>>>(T, ei, dinv, hB);
    bias_relu_kernel<<<aggGrid, blk256, 0, stream>>>(hB, b2, total);
    g_acc_kernel<<<gaccGrid, blk256, 0, stream>>>(hB, Wfc, g, M, 1);

    // --- layer 3: hB(K=128) -> hA (reuse) ---
    gemm_wmma_f32<<<gemmGrid, blk256, 0, stream>>>(hB, W3, T, M, 128);
    agg_self_kernel<<<aggGrid, blk256, 0, stream>>>(T, dinv, hA, total);
    agg_edge_kernel<<<edgeGrid, dim3(HFEAT), 0, stream>>>(T, ei, dinv, hA);
    bias_relu_kernel<<<aggGrid, blk256, 0, stream>>>(hA, b3, total);
    g_acc_kernel<<<gaccGrid, blk256, 0, stream>>>(hA, Wfc, g, M, 2);

    // --- readout ---
    z_init_kernel<<<dim3((BS * 512 + 255) / 256), blk256, 0, stream>>>(bl1, z);
    z_acc_kernel<<<dim3(BS, (NNODES + ZCHUNK - 1) / ZCHUNK), dim3(512), 0, stream>>>(g, Wl1, z);
    z_relu_kernel<<<dim3((BS * 512 + 255) / 256), blk256, 0, stream>>>(z);
    head_kernel<<<dim3(BS), dim3(512), 0, stream>>>(z, Wl2, bl2, out);
}